// CRFLayer_38671885533455
// MI455X (gfx1250) — compile-verified
//
#include <hip/hip_runtime.h>
#include <hip/hip_bf16.h>

// ---------------- problem constants ----------------
#define BB 4
#define HH 384
#define WW 384
#define CC 21
#define NPIX (BB * HH * WW)          // 589824
#define NELEM (NPIX * CC)            // 12386304
#define R_G 6                        // gaussian radius (sigma=3 -> ceil(2*3))
#define INV2TA2 (1.0f / 12800.0f)    // 1/(2*80^2)   bilateral spatial
#define INV2TB2 (1.0f / 338.0f)      // 1/(2*13^2)   bilateral color
#define INV2SP2 (1.0f / 8.0f)        // 1/(2*2^2)    jbu spatial
#define INV2SB2 (1.0f / 338.0f)      // 1/(2*13^2)   jbu color

// exp(-i^2/18) for i = -6..6
__constant__ float c_GK[13] = {
    0.135335f, 0.249352f, 0.411112f, 0.606531f, 0.800737f, 0.945960f, 1.0f,
    0.945960f, 0.800737f, 0.606531f, 0.411112f, 0.249352f, 0.135335f};

typedef float v2f __attribute__((ext_vector_type(2)));
typedef float v8f __attribute__((ext_vector_type(8)));

// ---------------- M1 = SW*CM, M2 = BW*CM, zero-padded to 24x32 ----------------
__global__ void combine_mats_kernel(const float* __restrict__ SW,
                                    const float* __restrict__ BW,
                                    const float* __restrict__ CM,
                                    float* __restrict__ M1p,
                                    float* __restrict__ M2p) {
  for (int idx = threadIdx.x; idx < 24 * 32; idx += blockDim.x) {
    int k = idx >> 5, n = idx & 31;
    float v1 = 0.f, v2 = 0.f;
    if (k < CC && n < CC) {
      for (int c = 0; c < CC; c++) {
        float cm = CM[c * CC + n];
        v1 += SW[k * CC + c] * cm;
        v2 += BW[k * CC + c] * cm;
      }
    }
    M1p[idx] = v1;
    M2p[idx] = v2;
  }
}

// ---------------- bilateral_norm = bilateral_filter(ones) - 1 (per pixel) ----
__global__ void bnorm_kernel(const float* __restrict__ img, float* __restrict__ bnorm) {
  int p = blockIdx.x * blockDim.x + threadIdx.x;
  if (p >= NPIX) return;
  int w = p % WW;
  int h = (p / WW) % HH;
  float c0 = img[p * 3], c1 = img[p * 3 + 1], c2 = img[p * 3 + 2];
  float s = 0.f;
  for (int dy = -2; dy <= 2; dy++) {
    int hh = h + dy;
    if (hh < 0 || hh >= HH) continue;
    for (int dx = -2; dx <= 2; dx++) {
      int ww = w + dx;
      if (ww < 0 || ww >= WW) continue;
      int pn = p + dy * WW + dx;
      float d0 = c0 - img[pn * 3], d1 = c1 - img[pn * 3 + 1], d2 = c2 - img[pn * 3 + 2];
      float sd = (float)(dy * dy + dx * dx);
      s += __expf(-(sd * INV2TA2 + (d0 * d0 + d1 * d1 + d2 * d2) * INV2TB2));
    }
  }
  bnorm[p] = s - 1.0f;
}

// ---------------- Q0 = softmax(-unary) ----------------
__global__ void init_softmax_kernel(const float* __restrict__ unary, float* __restrict__ Q) {
  int p = blockIdx.x * blockDim.x + threadIdx.x;
  if (p >= NPIX) return;
  float v[CC];
  float mx = -1e30f;
#pragma unroll
  for (int c = 0; c < CC; c++) {
    v[c] = -unary[p * CC + c];
    mx = fmaxf(mx, v[c]);
  }
  float sum = 0.f;
#pragma unroll
  for (int c = 0; c < CC; c++) {
    v[c] = __expf(v[c] - mx);
    sum += v[c];
  }
  float inv = 1.0f / sum;
#pragma unroll
  for (int c = 0; c < CC; c++) Q[p * CC + c] = v[c] * inv;
}

// ---------------- gaussian vertical pass ----------------
__global__ void gaussV_kernel(const float* __restrict__ in, float* __restrict__ out) {
  int t = blockIdx.x * blockDim.x + threadIdx.x;
  if (t >= NELEM) return;
  int rest = t / CC;
  int h = (rest / WW) % HH;
  float s = 0.f;
#pragma unroll
  for (int dy = -R_G; dy <= R_G; dy++) {
    int hhv = h + dy;
    if (hhv >= 0 && hhv < HH) s += c_GK[dy + R_G] * in[t + dy * (WW * CC)];
  }
  out[t] = s;
}

// ---------------- gaussian horizontal pass + (g - Q)/spatial_norm ------------
__global__ void gaussH_fin_kernel(const float* __restrict__ tmp, const float* __restrict__ Q,
                                  float* __restrict__ out) {
  int t = blockIdx.x * blockDim.x + threadIdx.x;
  if (t >= NELEM) return;
  int rest = t / CC;
  int w = rest % WW;
  int h = (rest / WW) % HH;
  float s = 0.f;
#pragma unroll
  for (int dx = -R_G; dx <= R_G; dx++) {
    int wwv = w + dx;
    if (wwv >= 0 && wwv < WW) s += c_GK[dx + R_G] * tmp[t + dx * CC];
  }
  float rc = 0.f, cc = 0.f;
#pragma unroll
  for (int i = 0; i < 13; i++) {
    int hhv = h + i - R_G;
    if (hhv >= 0 && hhv < HH) rc += c_GK[i];
    int wwv = w + i - R_G;
    if (wwv >= 0 && wwv < WW) cc += c_GK[i];
  }
  out[t] = (s - Q[t]) / (rc * cc - 1.0f);
}

// ---------------- bilateral filter minus center: out = bil(Q) - Q -----------
__global__ void bilateral_kernel(const float* __restrict__ Q, const float* __restrict__ img,
                                 float* __restrict__ out) {
  int p = blockIdx.x * blockDim.x + threadIdx.x;
  if (p >= NPIX) return;
  int w = p % WW;
  int h = (p / WW) % HH;
  float c0 = img[p * 3], c1 = img[p * 3 + 1], c2 = img[p * 3 + 2];
  float acc[CC];
#pragma unroll
  for (int c = 0; c < CC; c++) acc[c] = 0.f;
  for (int dy = -2; dy <= 2; dy++) {
    int hhv = h + dy;
    if (hhv < 0 || hhv >= HH) continue;
    for (int dx = -2; dx <= 2; dx++) {
      int wwv = w + dx;
      if (wwv < 0 || wwv >= WW) continue;
      int pn = p + dy * WW + dx;
      float d0 = c0 - img[pn * 3], d1 = c1 - img[pn * 3 + 1], d2 = c2 - img[pn * 3 + 2];
      float sd = (float)(dy * dy + dx * dx);
      float wgt = __expf(-(sd * INV2TA2 + (d0 * d0 + d1 * d1 + d2 * d2) * INV2TB2));
      const float* qn = Q + pn * CC;
#pragma unroll
      for (int c = 0; c < CC; c++) acc[c] += wgt * qn[c];
    }
  }
#pragma unroll
  for (int c = 0; c < CC; c++) out[p * CC + c] = acc[c] - Q[p * CC + c];
}

// ---------------- JBU filter, fused /bilateral_norm --------------------------
__global__ void jbu_kernel(const float* __restrict__ x, const float* __restrict__ img,
                           const float* __restrict__ bnorm, float* __restrict__ out) {
  int p = blockIdx.x * blockDim.x + threadIdx.x;
  if (p >= NPIX) return;
  int w = p % WW;
  int h = (p / WW) % HH;
  float c0 = img[p * 3], c1 = img[p * 3 + 1], c2 = img[p * 3 + 2];
  float num[CC];
  float den = 0.f;
#pragma unroll
  for (int c = 0; c < CC; c++) num[c] = 0.f;
  for (int dy = -2; dy <= 2; dy++) {
    int hhv = h + dy;
    if (hhv < 0 || hhv >= HH) continue;
    for (int dx = -2; dx <= 2; dx++) {
      int wwv = w + dx;
      if (wwv < 0 || wwv >= WW) continue;
      int pn = p + dy * WW + dx;
      float d0 = c0 - img[pn * 3], d1 = c1 - img[pn * 3 + 1], d2 = c2 - img[pn * 3 + 2];
      float sd = (float)(dy * dy + dx * dx);
      float wgt = __expf(-(sd * INV2SP2 + (d0 * d0 + d1 * d1 + d2 * d2) * INV2SB2));
      const float* xn = x + pn * CC;
#pragma unroll
      for (int c = 0; c < CC; c++) num[c] += wgt * xn[c];
      den += wgt;
    }
  }
  float inv = 1.0f / (fmaxf(den, 1e-20f) * bnorm[p]);
#pragma unroll
  for (int c = 0; c < CC; c++) out[p * CC + c] = num[c] * inv;
}

// ---------------- WMMA einsum chain + softmax update -------------------------
// pairwise = spat · M1p + jbu · M2p   (M = SW*CM, BW*CM, padded 24x32)
// Q = softmax(-unary - pairwise)
// One wave handles 16 pixels x 32 (padded) classes via V_WMMA_F32_16X16X4_F32.
__global__ void update_kernel(const float* __restrict__ spat, const float* __restrict__ jbu,
                              const float* __restrict__ unary, const float* __restrict__ M1p,
                              const float* __restrict__ M2p, float* __restrict__ Qout) {
  int wave = threadIdx.x >> 5;
  int lane = threadIdx.x & 31;
  int tile = blockIdx.x * 8 + wave;
  int p0 = tile * 16;
  int half = lane >> 4;  // 0: K 0-1 (A) / rows k,k+1 (B); 1: K 2-3 / rows k+2,k+3
  int ln = lane & 15;

  v8f acc0 = {};  // output classes n = 0..15
  v8f acc1 = {};  // output classes n = 16..31 (21..31 are padding)
#pragma unroll
  for (int ks = 0; ks < 6; ks++) {
    int kb = ks * 4;
    int cA = kb + half * 2;            // class index carried by this lane in A
    int rowp = (p0 + ln) * CC;         // pixel row for A (lanes 0-15 = M 0-15)
    v2f aS, aJ;
    aS.x = (cA < CC) ? spat[rowp + cA] : 0.f;
    aS.y = (cA + 1 < CC) ? spat[rowp + cA + 1] : 0.f;
    aJ.x = (cA < CC) ? jbu[rowp + cA] : 0.f;
    aJ.y = (cA + 1 < CC) ? jbu[rowp + cA + 1] : 0.f;
    int r0 = (kb + half * 2) * 32;     // B rows: vgpr0 -> k, k+2 ; vgpr1 -> k+1, k+3
    int r1 = (kb + 1 + half * 2) * 32;
    v2f b10, b11, b20, b21;
    b10.x = M1p[r0 + ln];       b10.y = M1p[r1 + ln];
    b11.x = M1p[r0 + 16 + ln];  b11.y = M1p[r1 + 16 + ln];
    b20.x = M2p[r0 + ln];       b20.y = M2p[r1 + ln];
    b21.x = M2p[r0 + 16 + ln];  b21.y = M2p[r1 + 16 + ln];
    acc0 = __builtin_amdgcn_wmma_f32_16x16x4_f32(false, aS, false, b10, (short)0, acc0, false, false);
    acc1 = __builtin_amdgcn_wmma_f32_16x16x4_f32(false, aS, false, b11, (short)0, acc1, false, false);
    acc0 = __builtin_amdgcn_wmma_f32_16x16x4_f32(false, aJ, false, b20, (short)0, acc0, false, false);
    acc1 = __builtin_amdgcn_wmma_f32_16x16x4_f32(false, aJ, false, b21, (short)0, acc1, false, false);
  }

  // softmax over n = 0..20 for each pixel row m; C/D layout:
  // vgpr r, lane l: n = l&15, m = r + (l>=16 ? 8 : 0)
#pragma unroll
  for (int r = 0; r < 8; r++) {
    int m = r + half * 8;
    int p = p0 + m;
    float x0 = -unary[p * CC + ln] - acc0[r];
    int n1 = 16 + ln;
    float x1 = (n1 < CC) ? (-unary[p * CC + n1] - acc1[r]) : -1e30f;
    float mx = fmaxf(x0, x1);
    for (int o = 1; o < 16; o <<= 1) mx = fmaxf(mx, __shfl_xor(mx, o, 16));
    float e0 = __expf(x0 - mx);
    float e1 = (n1 < CC) ? __expf(x1 - mx) : 0.f;
    float s = e0 + e1;
    for (int o = 1; o < 16; o <<= 1) s += __shfl_xor(s, o, 16);
    float inv = 1.0f / s;
    Qout[p * CC + ln] = e0 * inv;
    if (n1 < CC) Qout[p * CC + n1] = e1 * inv;
  }
}

// ---------------- launch ----------------
extern "C" void kernel_launch(void* const* d_in, const int* in_sizes, int n_in,
                              void* d_out, int out_size, void* d_ws, size_t ws_size,
                              hipStream_t stream) {
  const float* unary = (const float*)d_in[0];
  const float* image = (const float*)d_in[1];
  const float* SW = (const float*)d_in[2];
  const float* BW = (const float*)d_in[3];
  const float* CM = (const float*)d_in[4];

  float* Q = (float*)d_ws;
  float* bufA = Q + NELEM;        // gaussian tmp / bilateral out (reused)
  float* bufB = bufA + NELEM;     // spatial_out
  float* bufC = bufB + NELEM;     // jbu out
  float* bnorm = bufC + NELEM;
  float* M1p = bnorm + NPIX;
  float* M2p = M1p + 24 * 32;

  const int TPB = 256;
  dim3 blk(TPB);
  dim3 gPix((NPIX + TPB - 1) / TPB);     // 2304
  dim3 gElem((NELEM + TPB - 1) / TPB);   // 48384
  dim3 gUpd(NPIX / 16 / 8);              // 4608 (NPIX divisible by 128)

  combine_mats_kernel<<<1, TPB, 0, stream>>>(SW, BW, CM, M1p, M2p);
  bnorm_kernel<<<gPix, blk, 0, stream>>>(image, bnorm);
  init_softmax_kernel<<<gPix, blk, 0, stream>>>(unary, Q);

  for (int it = 0; it < 5; it++) {
    gaussV_kernel<<<gElem, blk, 0, stream>>>(Q, bufA);
    gaussH_fin_kernel<<<gElem, blk, 0, stream>>>(bufA, Q, bufB);
    bilateral_kernel<<<gPix, blk, 0, stream>>>(Q, image, bufA);
    jbu_kernel<<<gPix, blk, 0, stream>>>(bufA, image, bnorm, bufC);
    update_kernel<<<gUpd, blk, 0, stream>>>(bufB, bufC, unary, M1p, M2p, Q);
  }

  hipMemcpyAsync(d_out, Q, (size_t)NELEM * sizeof(float), hipMemcpyDeviceToDevice, stream);
}